// ScipyRotation_91250875171654
// MI455X (gfx1250) — compile-verified
//
#include <hip/hip_runtime.h>
#include <stdint.h>

// ---------------------------------------------------------------------------
// ScipyRotation (order-3 spline, 5 deg, constant boundary) + MSE/mask loss
// MI455X / gfx1250 implementation.
//
// Pipeline (all on `stream`):
//   1. iir_causal / iir_anticausal  (vertical)   : fm2 -> bufA -> bufB
//   2. iir_causal / iir_anticausal  (horizontal) : bufB -> bufA -> bufB
//      (16-way chunked recursion, 32-step warm-up; |z|^32 ~ 1e-18 << fp32 ulp)
//   3. rot_reduce : per 32x8 output tile, TDM-stage a 44x20 coef window
//      into LDS (tensor_load_to_lds + s_wait_tensorcnt), 4x4 cubic gather
//      from LDS, analytic mask (prefiltered ones == 1.0), block reduction.
//   4. finalize : deterministic fixed-order reduction -> loss scalar.
//
// Roofline: working set (fm1 + coef, 128 MB) fits the 192 MB L2; HBM traffic
// is ~128 MB (fm2 in, fm1 in) => ~5.5 us floor at 23.3 TB/s. The IIR passes
// are latency-bound dependent-FMA chains, hence the chunk parallelization.
// ---------------------------------------------------------------------------

#define NDIM   4096
#define ZP     (-0.26794919243112270647f)   // sqrt(3) - 2  (cubic B-spline pole)
#define GAIN   6.0f                          // (1-z)(1-1/z) == 6 exactly
#define DLASTF 0.28867513459481288225f      // z/(z^2-1) == 1/(2*sqrt(3))
#define WARM   32
#define NCHUNK 16                            // 4096/16 = 256 >> WARM
#define COSA   0.99619469809174553230f      // cos(5 deg), fp32
#define SINA   0.08715574274765817356f      // sin(5 deg), fp32
#define CTR    2047.5f                       // (4096-1)/2

#define OTX    32                            // output tile width  (one wave row)
#define OTY    8                             // output tile height (8 waves/block)
#define TILE_W 44                            // staged input window (>= 32*cos+8*sin+6)
#define TILE_H 20                            // staged input window (>=  8*cos+32*sin+6)

typedef unsigned int uint32;
typedef uint32 u32x4 __attribute__((ext_vector_type(4)));
typedef int    i32x4 __attribute__((ext_vector_type(4)));
typedef int    i32x8 __attribute__((ext_vector_type(8)));

// ---------------------------------------------------------------------------
// Chunk-parallel causal IIR pass:  c[n] = gain*x[n] + z*c[n-1]
// lineStride/elemStride select vertical (1, N) or horizontal (N, 1) filtering.
// Chunk 0 uses the exact 30-term truncated-mirror init (matches reference);
// interior chunks warm up WARM steps (error ~ z^WARM, far below fp32 ulp).
// ---------------------------------------------------------------------------
__global__ __launch_bounds__(256)
void iir_causal(const float* __restrict__ x, float* __restrict__ c,
                int N, int chunkLen, size_t lineStride, size_t elemStride) {
    int line = blockIdx.x * blockDim.x + threadIdx.x;
    size_t base = (size_t)line * lineStride;
    int r0 = blockIdx.y * chunkLen;
    int r1 = r0 + chunkLen;
    float cc;
    int n0;
    if (r0 == 0) {
        float acc = 0.0f, p = 1.0f;
        for (int k = 0; k < 30; ++k) {
            acc += p * (GAIN * x[base + (size_t)k * elemStride]);
            p *= ZP;
        }
        cc = acc;
        c[base] = cc;
        n0 = 1;
    } else {
        int rw = r0 - WARM;
        cc = GAIN * x[base + (size_t)rw * elemStride];
        for (int n = rw + 1; n < r0; ++n)
            cc = fmaf(ZP, cc, GAIN * x[base + (size_t)n * elemStride]);
        n0 = r0;
    }
#pragma unroll 4
    for (int n = n0; n < r1; ++n) {
        cc = fmaf(ZP, cc, GAIN * x[base + (size_t)n * elemStride]);
        c[base + (size_t)n * elemStride] = cc;
    }
}

// ---------------------------------------------------------------------------
// Chunk-parallel anticausal pass:  d[n] = z*(d[n+1] - c[n])
// Last chunk uses the exact mirror init; interior chunks warm up from zero
// over WARM steps using true c values (init error decays as z^WARM).
// Safe in-place (c==d) when chunkLen == N (single chunk).
// ---------------------------------------------------------------------------
__global__ __launch_bounds__(256)
void iir_anticausal(const float* __restrict__ c, float* __restrict__ d,
                    int N, int chunkLen, size_t lineStride, size_t elemStride) {
    int line = blockIdx.x * blockDim.x + threadIdx.x;
    size_t base = (size_t)line * lineStride;
    int r0 = blockIdx.y * chunkLen;
    int r1 = r0 + chunkLen;
    float dd;
    int start;
    if (r1 >= N) {
        float c1 = c[base + (size_t)(N - 1) * elemStride];
        float c2 = c[base + (size_t)(N - 2) * elemStride];
        dd = DLASTF * (c1 + ZP * c2);
        d[base + (size_t)(N - 1) * elemStride] = dd;
        start = N - 2;
    } else {
        dd = 0.0f;
        for (int n = r1 + WARM - 1; n >= r1; --n)
            dd = ZP * (dd - c[base + (size_t)n * elemStride]);
        start = r1 - 1;
    }
#pragma unroll 4
    for (int n = start; n >= r0; --n) {
        dd = ZP * (dd - c[base + (size_t)n * elemStride]);
        d[base + (size_t)n * elemStride] = dd;
    }
}

__device__ __forceinline__ void cubic_w(float f, float w[4]) {
    float f2 = f * f, f3 = f2 * f;
    w[0] = (1.0f - 3.0f * f + 3.0f * f2 - f3) * (1.0f / 6.0f);
    w[1] = (4.0f - 6.0f * f2 + 3.0f * f3) * (1.0f / 6.0f);
    w[2] = (1.0f + 3.0f * f + 3.0f * f2 - 3.0f * f3) * (1.0f / 6.0f);
    w[3] = f3 * (1.0f / 6.0f);
}

// ---------------------------------------------------------------------------
// Rotation + fused loss partials. One 32x8 output tile per 256-thread block.
// The needed coefficient window is staged into LDS by the Tensor Data Mover
// (one tensor_load_to_lds per block, issued by wave 0, TENSORcnt-waited).
// Mask term computed analytically: spline-prefiltered ones == 1.0 exactly,
// so r_mask(pixel) = sum of in-bounds cubic tap weights.
// ---------------------------------------------------------------------------
__global__ __launch_bounds__(256)
void rot_reduce(const float* __restrict__ coef, const float* __restrict__ fm1,
                float* __restrict__ partials) {
    __shared__ float tile[TILE_H][TILE_W];
    __shared__ float sE[256];
    __shared__ float sM[256];
    const int H = NDIM, W = NDIM;
    int ox0 = blockIdx.x * OTX;
    int oy0 = blockIdx.y * OTY;

    // Block-uniform input-region bounds from the 4 output-tile corners
    // (inverse rotation is affine -> extrema at corners).
    float iymin = 1e30f, iymax = -1e30f, ixmin = 1e30f, ixmax = -1e30f;
    for (int j = 0; j < 2; ++j) {
        float yv = (float)(oy0 + j * (OTY - 1)) - CTR;
        for (int i = 0; i < 2; ++i) {
            float xv = (float)(ox0 + i * (OTX - 1)) - CTR;
            float iy = COSA * yv - SINA * xv + CTR;
            float ix = SINA * yv + COSA * xv + CTR;
            iymin = fminf(iymin, iy); iymax = fmaxf(iymax, iy);
            ixmin = fminf(ixmin, ix); ixmax = fmaxf(ixmax, ix);
        }
    }
    int y0 = (int)floorf(iymin) - 2; y0 = y0 < 0 ? 0 : (y0 > H - 1 ? H - 1 : y0);
    int x0 = (int)floorf(ixmin) - 2; x0 = x0 < 0 ? 0 : (x0 > W - 1 ? W - 1 : x0);
    y0 = __builtin_amdgcn_readfirstlane(y0);
    x0 = __builtin_amdgcn_readfirstlane(x0);

    if (threadIdx.y == 0) {
        // ---- Tensor Data Mover: stage TILE_H x TILE_W f32 window into LDS ----
        // D# group0: count=1 | lds_addr | global_addr[56:0] | type=2
        // D# group1: data_size=4B, tensor_dim = remaining extent (right/bottom
        //            OOB zero-fills the LDS tile), tile dims, row stride = W.
        uint32 lds = (uint32)(uint64_t)(uintptr_t)(&tile[0][0]); // low 32b of LDS aperture addr == LDS byte offset
        uint64_t ga = (uint64_t)(uintptr_t)coef + 4ull * ((uint64_t)y0 * (uint64_t)W + (uint64_t)x0);
        u32x4 g0;
        g0[0] = 1u;                                            // count = 1 (valid user D#)
        g0[1] = lds;                                           // lds_addr
        g0[2] = (uint32)ga;                                    // global_addr[31:0]
        g0[3] = (uint32)((ga >> 32) & 0x01FFFFFFu) | (2u << 30); // addr[56:32] | type=2
        i32x8 g1;
        g1[0] = (int)(2u << 16);                               // data_size = 2 (4 bytes)
        g1[1] = (int)(((uint32)(W - x0) & 0xFFFFu) << 16);     // tensor_dim0[15:0] @ bits 63:48
        g1[2] = (int)(((uint32)(H - y0) & 0xFFFFu) << 16);     // tensor_dim1[15:0] @ bits 95:80
        g1[3] = (int)(((uint32)TILE_W) << 16);                 // tile_dim0 @ bits 127:112
        g1[4] = (int)TILE_H;                                   // tile_dim1 @ bits 143:128
        g1[5] = (int)W;                                        // tensor_dim0_stride[31:0]
        g1[6] = 0;
        g1[7] = 0;
        i32x4 gz; gz[0] = 0; gz[1] = 0; gz[2] = 0; gz[3] = 0;  // 2D tile: groups 2/3 unused
#if __has_include(<hip/amd_detail/amd_gfx1250_TDM.h>)
        i32x8 gz8; gz8[0]=0; gz8[1]=0; gz8[2]=0; gz8[3]=0; gz8[4]=0; gz8[5]=0; gz8[6]=0; gz8[7]=0;
        __builtin_amdgcn_tensor_load_to_lds(g0, g1, gz, gz, gz8, 0);   // clang-23 / therock: 6 args
#else
        __builtin_amdgcn_tensor_load_to_lds(g0, g1, gz, gz, 0);        // ROCm 7.2 / clang-22: 5 args
#endif
        __builtin_amdgcn_s_wait_tensorcnt(0);
    }
    __syncthreads();

    int x = ox0 + (int)threadIdx.x;
    int y = oy0 + (int)threadIdx.y;
    float yv = (float)y - CTR, xv = (float)x - CTR;
    float iy = COSA * yv - SINA * xv + CTR;
    float ix = SINA * yv + COSA * xv + CTR;
    float by = floorf(iy), bx = floorf(ix);
    int byi = (int)by, bxi = (int)bx;
    float wy[4], wx[4];
    cubic_w(iy - by, wy);
    cubic_w(ix - bx, wx);

    float r = 0.0f, m = 0.0f;
#pragma unroll
    for (int dy = 0; dy < 4; ++dy) {
        int yi = byi + dy - 1;
        float okY = (yi >= 0 && yi < H) ? 1.0f : 0.0f;
        int ty = yi - y0; ty = ty < 0 ? 0 : (ty > TILE_H - 1 ? TILE_H - 1 : ty);
#pragma unroll
        for (int dx = 0; dx < 4; ++dx) {
            int xi = bxi + dx - 1;
            float ok = okY * ((xi >= 0 && xi < W) ? 1.0f : 0.0f);
            int tx = xi - x0; tx = tx < 0 ? 0 : (tx > TILE_W - 1 ? TILE_W - 1 : tx);
            float w = wy[dy] * wx[dx] * ok;
            r += w * tile[ty][tx];   // coef gather from LDS-staged window
            m += w;                  // analytic r_mask (coef(ones) == 1.0)
        }
    }
    float a = fm1[(size_t)y * W + x];
    float e = a - r;

    int t = threadIdx.y * OTX + threadIdx.x;
    sE[t] = e * e;
    sM[t] = m;
    __syncthreads();
    for (int s = 128; s > 0; s >>= 1) {
        if (t < s) { sE[t] += sE[t + s]; sM[t] += sM[t + s]; }
        __syncthreads();
    }
    if (t == 0) {
        int bid = blockIdx.y * gridDim.x + blockIdx.x;
        partials[2 * bid]     = sE[0];
        partials[2 * bid + 1] = sM[0];
    }
}

// Deterministic fixed-order final reduction -> loss scalar.
__global__ __launch_bounds__(256)
void finalize_loss(const float* __restrict__ partials, int nblocks,
                   float* __restrict__ out) {
    __shared__ float sE[256];
    __shared__ float sM[256];
    int t = threadIdx.x;
    float e = 0.0f, m = 0.0f;
    for (int i = t; i < nblocks; i += 256) {
        e += partials[2 * i];
        m += partials[2 * i + 1];
    }
    sE[t] = e; sM[t] = m;
    __syncthreads();
    for (int s = 128; s > 0; s >>= 1) {
        if (t < s) { sE[t] += sE[t + s]; sM[t] += sM[t + s]; }
        __syncthreads();
    }
    if (t == 0) out[0] = sE[0] / sM[0];
}

extern "C" void kernel_launch(void* const* d_in, const int* in_sizes, int n_in,
                              void* d_out, int out_size, void* d_ws, size_t ws_size,
                              hipStream_t stream) {
    (void)in_sizes; (void)n_in; (void)out_size;
    const int N = NDIM;
    const float* fm1 = (const float*)d_in[0];
    const float* fm2 = (const float*)d_in[1];
    float* out = (float*)d_out;

    size_t img = (size_t)N * N * sizeof(float);
    int nrb = (N / OTX) * (N / OTY);                 // rotation blocks
    size_t partBytes = (size_t)nrb * 2 * sizeof(float);

    char* ws = (char*)d_ws;
    // Prefer chunk-parallel ping-pong (needs 2 image buffers); fall back to
    // the exact single-chunk in-place path if the workspace is small.
    bool chunked = ws_size >= 2 * img + partBytes;
    float* bufA = (float*)ws;
    float* bufB = chunked ? (float*)(ws + img) : bufA;
    float* partials = (float*)(ws + (chunked ? 2 * img : img));
    int nch = chunked ? NCHUNK : 1;
    int chunkLen = N / nch;

    dim3 ib(256, 1, 1), ig(N / 256, nch, 1);
    // vertical prefilter (axis 0): fm2 -> bufA -> bufB
    iir_causal<<<ig, ib, 0, stream>>>(fm2, bufA, N, chunkLen, (size_t)1, (size_t)N);
    iir_anticausal<<<ig, ib, 0, stream>>>(bufA, bufB, N, chunkLen, (size_t)1, (size_t)N);
    // horizontal prefilter (axis 1): bufB -> bufA -> bufB  (L2-resident; 192MB L2)
    iir_causal<<<ig, ib, 0, stream>>>(bufB, bufA, N, chunkLen, (size_t)N, (size_t)1);
    iir_anticausal<<<ig, ib, 0, stream>>>(bufA, bufB, N, chunkLen, (size_t)N, (size_t)1);

    dim3 rb(OTX, OTY, 1), rg(N / OTX, N / OTY, 1);
    rot_reduce<<<rg, rb, 0, stream>>>(bufB, fm1, partials);
    finalize_loss<<<1, 256, 0, stream>>>(partials, nrb, out);
}